// get_graph_feature_15023795602176
// MI455X (gfx1250) — compile-verified
//
#include <hip/hip_runtime.h>
#include <math.h>

typedef __attribute__((ext_vector_type(2))) float v2f;
typedef __attribute__((ext_vector_type(8))) float v8f;

#define B_ 16
#define C_ 64
#define N_ 2048
#define K_ 20
#define EPS_ 1e-5f

#define NSTRIPS (B_ * (N_ / 16))      // 2048 strips of 16 rows
#define STATS_BLOCKS 1024

// ---------------------------------------------------------------------------
// xx[b*N + n] = sum_c x[b,c,n]^2
// ---------------------------------------------------------------------------
__global__ __launch_bounds__(256) void xx_kernel(const float* __restrict__ x,
                                                 float* __restrict__ xx) {
    int t = blockIdx.x * blockDim.x + threadIdx.x;
    if (t >= B_ * N_) return;
    int b = t / N_;
    int n = t % N_;
    const float* xb = x + (size_t)b * C_ * N_ + n;
    float s = 0.f;
#pragma unroll
    for (int c = 0; c < C_; ++c) {
        float v = xb[(size_t)c * N_];
        s += v * v;
    }
    xx[t] = s;
}

// ---------------------------------------------------------------------------
// Fused Gram (f32 WMMA) + top-K selection.
// 8 waves per block; each wave owns one 16-row strip of one batch and sweeps
// all 128 column tiles, maintaining per-row sorted top-20 lists in LDS.
// pdist[n][m] = 2*S[n][m] - xx[n] - xx[m]
// ---------------------------------------------------------------------------
__global__ __launch_bounds__(256) void gram_topk_kernel(const float* __restrict__ x,
                                                        const float* __restrict__ xx,
                                                        int* __restrict__ idx_out) {
    const int lane = threadIdx.x & 31;
    const int wave = threadIdx.x >> 5;
    const int strip = blockIdx.x * 8 + wave;       // 0 .. NSTRIPS-1 (grid sized exactly)
    const int b  = strip / (N_ / 16);
    const int n0 = (strip % (N_ / 16)) * 16;

    const float* xb  = x  + (size_t)b * C_ * N_;
    const float* xxb = xx + b * N_;

    const int row = lane & 15;       // M (for A) / N-col (for B/D)
    const int hi  = lane >> 4;       // which K half this lane holds

    // Preload A strip: 16 K-chunks of 4; lane holds (c, c+1) with c = 4*kc + 2*hi
    v2f a[16];
#pragma unroll
    for (int kc = 0; kc < 16; ++kc) {
        int c = 4 * kc + 2 * hi;
        a[kc].x = xb[(size_t)c       * N_ + n0 + row];
        a[kc].y = xb[(size_t)(c + 1) * N_ + n0 + row];
    }

    // xx for the 8 row-values this lane's accumulator holds (M = i + 8*hi)
    float xxrow[8];
#pragma unroll
    for (int i = 0; i < 8; ++i) xxrow[i] = xxb[n0 + i + 8 * hi];

    __shared__ float tile [8][16][16];
    __shared__ float tvals[8][16][K_];
    __shared__ int   tidxs[8][16][K_];

    if (lane < 16) {
        for (int j = 0; j < K_; ++j) {
            tvals[wave][lane][j] = -3.4e38f;
            tidxs[wave][lane][j] = 0;
        }
    }
    __syncthreads();

    for (int m0 = 0; m0 < N_; m0 += 16) {
        v8f acc = {};
#pragma unroll
        for (int kc = 0; kc < 16; ++kc) {
            int c = 4 * kc + 2 * hi;
            v2f bb;
            bb.x = xb[(size_t)c       * N_ + m0 + row];
            bb.y = xb[(size_t)(c + 1) * N_ + m0 + row];
            // D = A x B + C,  f32 16x16x4
            acc = __builtin_amdgcn_wmma_f32_16x16x4_f32(
                false, a[kc], false, bb, (short)0, acc, false, false);
        }
        float xxm = xxb[m0 + row];
#pragma unroll
        for (int i = 0; i < 8; ++i) {
            float pd = 2.0f * acc[i] - xxrow[i] - xxm;
            tile[wave][i + 8 * hi][row] = pd;
        }
        __syncthreads();
        if (lane < 16) {
            float* vl = tvals[wave][lane];
            int*   il = tidxs[wave][lane];
#pragma unroll 1
            for (int j = 0; j < 16; ++j) {
                float v = tile[wave][lane][j];
                if (v > vl[K_ - 1]) {           // strict: equal keeps earlier (lower) index
                    int p = K_ - 1;
                    while (p > 0 && v > vl[p - 1]) {
                        vl[p] = vl[p - 1];
                        il[p] = il[p - 1];
                        --p;
                    }
                    vl[p] = v;
                    il[p] = m0 + j;
                }
            }
        }
        __syncthreads();
    }

    if (lane < 16) {
        int n = n0 + lane;
        for (int j = 0; j < K_; ++j)
            idx_out[(b * N_ + n) * K_ + j] = tidxs[wave][lane][j];
    }
}

// ---------------------------------------------------------------------------
// Deterministic two-stage reduction of sum(diff), sum(diff^2)
// ---------------------------------------------------------------------------
__global__ __launch_bounds__(256) void stats_partial_kernel(const float* __restrict__ x,
                                                            const int* __restrict__ idx,
                                                            float* __restrict__ part) {
    const long long total = (long long)B_ * C_ * N_ * K_;
    float s = 0.f, s2 = 0.f;
    for (long long t = (long long)blockIdx.x * blockDim.x + threadIdx.x; t < total;
         t += (long long)gridDim.x * blockDim.x) {
        int k = (int)(t % K_);
        long long r = t / K_;
        int n = (int)(r % N_); r /= N_;
        int c = (int)(r % C_); r /= C_;
        int b = (int)r;
        int m = idx[(b * N_ + n) * K_ + k];
        const float* xbc = x + (size_t)b * C_ * N_ + (size_t)c * N_;
        float d = xbc[m] - xbc[n];
        s += d;
        s2 += d * d;
    }
    __shared__ float rs[256], rs2[256];
    rs[threadIdx.x] = s;
    rs2[threadIdx.x] = s2;
    __syncthreads();
    for (int off = 128; off > 0; off >>= 1) {
        if ((int)threadIdx.x < off) {
            rs[threadIdx.x]  += rs[threadIdx.x + off];
            rs2[threadIdx.x] += rs2[threadIdx.x + off];
        }
        __syncthreads();
    }
    if (threadIdx.x == 0) {
        part[blockIdx.x] = rs[0];
        part[STATS_BLOCKS + blockIdx.x] = rs2[0];
    }
}

__global__ __launch_bounds__(256) void stats_final_kernel(const float* __restrict__ part,
                                                          float* __restrict__ stats) {
    int t = threadIdx.x;
    float s  = part[t] + part[t + 256] + part[t + 512] + part[t + 768];
    float s2 = part[STATS_BLOCKS + t] + part[STATS_BLOCKS + t + 256] +
               part[STATS_BLOCKS + t + 512] + part[STATS_BLOCKS + t + 768];
    __shared__ float rs[256], rs2[256];
    rs[t] = s; rs2[t] = s2;
    __syncthreads();
    for (int off = 128; off > 0; off >>= 1) {
        if (t < off) { rs[t] += rs[t + off]; rs2[t] += rs2[t + off]; }
        __syncthreads();
    }
    if (t == 0) { stats[0] = rs[0]; stats[1] = rs2[0]; }
}

// ---------------------------------------------------------------------------
// Epilogue: out[b, c, n, k]      = alpha[c] * (diff / (std+eps)) + beta[c]
//           out[b, C+c, n, k]    = x[b, c, n]
// ---------------------------------------------------------------------------
__global__ __launch_bounds__(256) void output_kernel(const float* __restrict__ x,
                                                     const int* __restrict__ idx,
                                                     const float* __restrict__ stats,
                                                     const float* __restrict__ alpha,
                                                     const float* __restrict__ beta,
                                                     float* __restrict__ out) {
    const long long total = (long long)B_ * C_ * N_ * K_;
    long long t = (long long)blockIdx.x * blockDim.x + threadIdx.x;
    if (t >= total) return;
    int k = (int)(t % K_);
    long long r = t / K_;
    int n = (int)(r % N_); r /= N_;
    int c = (int)(r % C_); r /= C_;
    int b = (int)r;

    const double cnt = (double)total;
    float sum = stats[0], sumsq = stats[1];
    float mean = (float)((double)sum / cnt);
    float var  = (float)(((double)sumsq - (double)sum * (double)mean) / (cnt - 1.0));
    float stdv = sqrtf(var);

    int m = idx[(b * N_ + n) * K_ + k];
    const float* xbc = x + (size_t)b * C_ * N_ + (size_t)c * N_;
    float xc = xbc[n];
    float d  = xbc[m] - xc;
    float feat = alpha[c] * (d / (stdv + EPS_)) + beta[c];

    size_t obase = (size_t)b * (2 * C_) * N_ * K_;
    out[obase + ((size_t)c * N_ + n) * (size_t)K_ + k] = feat;
    out[obase + ((size_t)(C_ + c) * N_ + n) * (size_t)K_ + k] = xc;
}

// ---------------------------------------------------------------------------
extern "C" void kernel_launch(void* const* d_in, const int* in_sizes, int n_in,
                              void* d_out, int out_size, void* d_ws, size_t ws_size,
                              hipStream_t stream) {
    const float* x     = (const float*)d_in[0];
    const float* alpha = (const float*)d_in[1];
    const float* beta  = (const float*)d_in[2];
    float* out = (float*)d_out;

    char* ws = (char*)d_ws;
    float* xx    = (float*)ws;                                   // B*N floats
    size_t off   = (size_t)B_ * N_ * sizeof(float);
    int*   idx   = (int*)(ws + off);                             // B*N*K ints
    off += (size_t)B_ * N_ * K_ * sizeof(int);
    float* part  = (float*)(ws + off);                           // 2*STATS_BLOCKS floats
    off += (size_t)2 * STATS_BLOCKS * sizeof(float);
    float* stats = (float*)(ws + off);                           // 2 floats

    xx_kernel<<<(B_ * N_ + 255) / 256, 256, 0, stream>>>(x, xx);
    gram_topk_kernel<<<NSTRIPS / 8, 256, 0, stream>>>(x, xx, idx);
    stats_partial_kernel<<<STATS_BLOCKS, 256, 0, stream>>>(x, idx, part);
    stats_final_kernel<<<1, 256, 0, stream>>>(part, stats);

    long long total = (long long)B_ * C_ * N_ * K_;
    int blocks = (int)((total + 255) / 256);
    output_kernel<<<blocks, 256, 0, stream>>>(x, idx, stats, alpha, beta, out);
}